// FullyNonlinearKANRNN_32349693674171
// MI455X (gfx1250) — compile-verified
//
#include <hip/hip_runtime.h>

typedef __attribute__((ext_vector_type(2))) float v2f;
typedef __attribute__((ext_vector_type(8))) float v8f;

#define HIDDEN   64
#define SEQL     96
#define NB       10
#define NF       (HIDDEN * NB)   // 640 features (K of the GEMM)
#define MBLK     64              // batch rows per block
#define NTHREADS 128             // 4 wave32 waves
#define NPAIR    (NF / 2)        // 320 K-pairs
#define BPSTRIDE 160             // floats per K-pair row: 128 data + 32 pad (stride%64==32 -> no bank clash)

// LDS carve (floats)
#define SMEM_FLOATS (NPAIR*BPSTRIDE + MBLK*HIDDEN + NF + NF + HIDDEN*2 + MBLK*NB + HIDDEN*NB)

// Branch-free tanh: prefer CDNA5 hardware V_TANH_F32 (TRANS32 pipe, co-executes
// with WMMA); fall back to exp2/rcp TRANS ops; never the branchy libm path.
__device__ __forceinline__ float fast_tanh(float v) {
#if __has_builtin(__builtin_amdgcn_tanhf)
    return __builtin_amdgcn_tanhf(v);
#elif __has_builtin(__builtin_amdgcn_exp2f) && __has_builtin(__builtin_amdgcn_rcpf)
    float e = __builtin_amdgcn_exp2f(v * 2.8853900817779268f);   // 2*log2(e)
    return 1.0f - 2.0f * __builtin_amdgcn_rcpf(e + 1.0f);
#else
    return tanhf(v);
#endif
}

__global__ __launch_bounds__(NTHREADS)
void kan_rnn_fused(const float* __restrict__ x,
                   const float* __restrict__ Wx,
                   const float* __restrict__ ax,
                   const float* __restrict__ cx,
                   const float* __restrict__ Wc,
                   const float* __restrict__ ac,
                   const float* __restrict__ cc,
                   const float* __restrict__ Wout,
                   float* __restrict__ out)
{
    extern __shared__ float smem[];
    float* sBp = smem;                          // [j][o*2 + par] = Wc[i][o][k], f=i*10+k, j=f/2, par=f&1
    float* sH  = sBp + NPAIR * BPSTRIDE;        // MBLK x HIDDEN (reused for phi)
    float* sAC = sH  + MBLK * HIDDEN;           // NF   -- keep sAC/sCC adjacent: compiler fuses the
    float* sCC = sAC + NF;                      // NF      pair into ds_load_2addr_stride64_b32
    float* sWo = sCC + NF;                      // HIDDEN*2
    float* sXT = sWo + HIDDEN * 2;              // MBLK*NB
    float* sWx = sXT + MBLK * NB;               // HIDDEN*NB

    const int tid = threadIdx.x;
    const int b0  = blockIdx.x * MBLK;

    __builtin_prefetch(Wc, 0, 1);

    // ---- Phase 0: stage weights into LDS (b128 coalesced reads of Wc) ----
    {
        const float4* Wc4 = (const float4*)Wc;
        for (int q = tid; q < (NF * HIDDEN) / 4; q += NTHREADS) {
            float4 v = Wc4[q];
            float vals[4] = { v.x, v.y, v.z, v.w };
            int w = q * 4;
#pragma unroll
            for (int e = 0; e < 4; ++e) {
                int ww = w + e;
                int i = ww / (HIDDEN * NB);
                int r = ww % (HIDDEN * NB);
                int o = r / NB;
                int k = r % NB;
                int f = i * NB + k;
                sBp[(f >> 1) * BPSTRIDE + o * 2 + (f & 1)] = vals[e];
            }
        }
    }
    for (int f = tid; f < NF; f += NTHREADS) { sAC[f] = ac[f]; sCC[f] = cc[f]; sWx[f] = Wx[f]; }
    for (int w = tid; w < HIDDEN * 2; w += NTHREADS) sWo[w] = Wout[w];

    // ---- Phase 1a: input basis on the ONLY column that matters: x[:, 95] ----
    // (step() returns comb[:, :HIDDEN] == tanh(x_phi): the recurrence is dead code)
    for (int idx = tid; idx < MBLK * NB; idx += NTHREADS) {
        int bl = idx / NB, k = idx % NB;
        float xv = x[(size_t)(b0 + bl) * SEQL + (SEQL - 1)];
        sXT[idx] = fast_tanh(ax[k] * xv + cx[k]);
    }
    __syncthreads();

    // ---- Phase 1b: h = tanh(x_phi) ----
    for (int idx = tid; idx < MBLK * HIDDEN; idx += NTHREADS) {
        int bl = idx / HIDDEN, o = idx % HIDDEN;
        float s = 0.f;
#pragma unroll
        for (int k = 0; k < NB; ++k) s += sWx[o * NB + k] * sXT[bl * NB + k];
        sH[idx] = fast_tanh(s);
    }
    __syncthreads();

    // ---- Phase 2: G = PhiFeat(64x640) x WcT(640x64) via V_WMMA_F32_16X16X4_F32 ----
    const int wave  = tid >> 5;
    const int lane  = tid & 31;
    const int m0    = wave * 16;           // local row base of this wave's strip
    const int rl    = m0 + (lane & 15);    // A-matrix row for this lane
    const int kh    = (lane >> 4) * 2;     // K sub-offset per ISA A 16x4 layout
    const int nl    = lane & 15;           // B/D column within tile
    const int jlane = lane >> 4;           // which K-pair of the 4-wide K step

    v8f acc0 = {}, acc1 = {}, acc2 = {}, acc3 = {};
    const float* hrow = &sH[rl * HIDDEN];

#pragma unroll 2
    for (int kk = 0; kk < NF; kk += 4) {
        const int f0 = kk + kh, f1 = f0 + 1;
        // A fragment: features on the fly; f flat-indexes ac/cc; tanh on TRANS pipe
        v2f A;
        A.x = fast_tanh(sAC[f0] * hrow[f0 / NB] + sCC[f0]);
        A.y = fast_tanh(sAC[f1] * hrow[f1 / NB] + sCC[f1]);

        // B fragments: one aligned ds_load_b64 per tile, no packing movs
        const float* bp = &sBp[((kk >> 1) + jlane) * BPSTRIDE + nl * 2];
        v2f B0 = *(const v2f*)(bp +  0);
        v2f B1 = *(const v2f*)(bp + 32);
        v2f B2 = *(const v2f*)(bp + 64);
        v2f B3 = *(const v2f*)(bp + 96);

        acc0 = __builtin_amdgcn_wmma_f32_16x16x4_f32(false, A, false, B0, (short)0, acc0, false, false);
        acc1 = __builtin_amdgcn_wmma_f32_16x16x4_f32(false, A, false, B1, (short)0, acc1, false, false);
        acc2 = __builtin_amdgcn_wmma_f32_16x16x4_f32(false, A, false, B2, (short)0, acc2, false, false);
        acc3 = __builtin_amdgcn_wmma_f32_16x16x4_f32(false, A, false, B3, (short)0, acc3, false, false);
    }
    __syncthreads();   // everyone done reading sH; safe to reuse it for phi

    // ---- Phase 3: phi = tanh(G) -> LDS (C/D layout: VGPR r -> row r + 8*(lane/16)) ----
#pragma unroll
    for (int r = 0; r < 8; ++r) {
        int row = m0 + r + 8 * (lane >> 4);
        sH[row * HIDDEN +  0 + nl] = fast_tanh(acc0[r]);
        sH[row * HIDDEN + 16 + nl] = fast_tanh(acc1[r]);
        sH[row * HIDDEN + 32 + nl] = fast_tanh(acc2[r]);
        sH[row * HIDDEN + 48 + nl] = fast_tanh(acc3[r]);
    }
    __syncthreads();

    // ---- Phase 4: head: out[b, c] = phi[b, :] . Wout[:, c]  (128 outputs, 1/thread) ----
    {
        const int bl = tid >> 1, c = tid & 1;
        float s = 0.f;
#pragma unroll 8
        for (int o = 0; o < HIDDEN; ++o) s += sH[bl * HIDDEN + o] * sWo[o * 2 + c];
        out[(size_t)(b0 + bl) * 2 + c] = s;
    }
}

extern "C" void kernel_launch(void* const* d_in, const int* in_sizes, int n_in,
                              void* d_out, int out_size, void* d_ws, size_t ws_size,
                              hipStream_t stream) {
    const float* x    = (const float*)d_in[0];
    const float* Wx   = (const float*)d_in[1];
    const float* ax   = (const float*)d_in[2];
    const float* cx   = (const float*)d_in[3];
    // d_in[4..6] = Wh, ah, ch: provably dead in the reference
    const float* Wc   = (const float*)d_in[7];
    const float* ac   = (const float*)d_in[8];
    const float* cc   = (const float*)d_in[9];
    const float* Wout = (const float*)d_in[10];
    float* out = (float*)d_out;

    const int Bsz = in_sizes[0] / SEQL;                       // 16384
    const size_t smem = (size_t)SMEM_FLOATS * sizeof(float);  // ~226 KB (< 320 KB/WGP)

    hipFuncSetAttribute((const void*)kan_rnn_fused,
                        hipFuncAttributeMaxDynamicSharedMemorySize, (int)smem);

    hipLaunchKernelGGL(kan_rnn_fused, dim3(Bsz / MBLK), dim3(NTHREADS), smem, stream,
                       x, Wx, ax, cx, Wc, ac, cc, Wout, out);
}